// GCNLayer_18202071400722
// MI455X (gfx1250) — compile-verified
//
#include <hip/hip_runtime.h>

typedef __attribute__((ext_vector_type(2))) float v2f;
typedef __attribute__((ext_vector_type(8))) float v8f;

#define IN_DIM  128
#define OUT_DIM 64

// ---------------------------------------------------------------------------
// Kernel 1: hs[n,:] = (X[n,:] @ W) * rsqrt(degree[n])
// Block = 256 threads = 8 waves. Block computes a 32x64 output tile:
//   wave (msub,nsub) with msub in {0,1}, nsub in {0..3} does a 16x16 tile
//   via 32 x V_WMMA_F32_16X16X4_F32 accumulations over K=128.
// LDS: A-tile 32x128 f32 (16 KB) + full W 128x64 f32 (32 KB) = 48 KB.
// ---------------------------------------------------------------------------
__global__ __launch_bounds__(256) void gcn_gemm_wmma(
    const float* __restrict__ X, const float* __restrict__ W,
    const float* __restrict__ degree, float* __restrict__ HS, int n_nodes)
{
    __shared__ float lA[32 * IN_DIM];        // 16 KB
    __shared__ float lW[IN_DIM * OUT_DIM];   // 32 KB

    const int tid = threadIdx.x;
    const int m0  = blockIdx.x * 32;

    // Stage W (128x64 = 8192 floats): 8 x float4 per thread, coalesced.
#pragma unroll
    for (int t = 0; t < 8; ++t) {
        const int idx = tid * 4 + t * 1024;
        *(float4*)(&lW[idx]) = *(const float4*)(W + idx);
    }
    // Stage A tile (32x128 = 4096 floats): 4 x float4 per thread, coalesced.
#pragma unroll
    for (int t = 0; t < 4; ++t) {
        const int idx = tid * 4 + t * 1024;
        const int row = m0 + idx / IN_DIM;
        float4 v = make_float4(0.f, 0.f, 0.f, 0.f);
        if (row < n_nodes)
            v = *(const float4*)(X + (size_t)m0 * IN_DIM + idx);
        *(float4*)(&lA[idx]) = v;
    }
    __syncthreads();

    const int lane = tid & 31;
    const int wid  = tid >> 5;
    const int msub = wid >> 2;      // 0..1 : M sub-tile
    const int nsub = wid & 3;       // 0..3 : N sub-tile
    const int half = lane >> 4;     // 0..1 : K-half per ISA 32-bit A/B layout
    const int l    = lane & 15;

    const int arow = msub * 16 + l;  // A row this lane feeds
    const int bcol = nsub * 16 + l;  // B col this lane feeds

    v8f acc = {0.f, 0.f, 0.f, 0.f, 0.f, 0.f, 0.f, 0.f};

#pragma unroll
    for (int k0 = 0; k0 < IN_DIM; k0 += 4) {
        const int ks = k0 + half * 2;           // lanes 0-15: K=k0,k0+1 ; 16-31: K=k0+2,k0+3
        v2f a = *(const v2f*)(&lA[arow * IN_DIM + ks]);   // ds_load_b64 (8B aligned)
        v2f b;
        b.x = lW[ks * OUT_DIM + bcol];
        b.y = lW[(ks + 1) * OUT_DIM + bcol];
        acc = __builtin_amdgcn_wmma_f32_16x16x4_f32(
            /*neg_a=*/false, a, /*neg_b=*/false, b,
            /*c_mod=*/(short)0, acc, /*reuse_a=*/false, /*reuse_b=*/false);
    }

    // Epilogue: fold in rsqrt(degree) per output row, store scaled h.
    // C/D layout: VGPR i -> M = half*8 + i (within 16-row tile), N = l.
#pragma unroll
    for (int i = 0; i < 8; ++i) {
        const int row = m0 + msub * 16 + half * 8 + i;
        if (row < n_nodes) {
            const float nrm = rsqrtf(degree[row]);
            HS[(size_t)row * OUT_DIM + nsub * 16 + l] = acc[i] * nrm;
        }
    }
}

// ---------------------------------------------------------------------------
// Kernel 2: for each edge e: out[dst[e], :] += hs[src[e], :]
// 16 threads per edge, float4 per thread -> coalesced 256B row gather,
// hardware fp32 atomics (global_atomic_add_f32) into L2-resident out.
// ---------------------------------------------------------------------------
__global__ __launch_bounds__(256) void gcn_edge_scatter(
    const float* __restrict__ HS, const int* __restrict__ src,
    const int* __restrict__ dst, float* __restrict__ out, int n_edges)
{
    const long long i = (long long)blockIdx.x * blockDim.x + threadIdx.x;
    const long long nwork = (long long)n_edges * 16;
    if (i >= nwork) return;

    const int e = (int)(i >> 4);
    const int q = (int)(i & 15);

    const int s = src[e];
    const int d = dst[e];

    const float4 v = *(const float4*)(HS + (size_t)s * OUT_DIM + q * 4);
    float* o = out + (size_t)d * OUT_DIM + q * 4;

    unsafeAtomicAdd(o + 0, v.x);
    unsafeAtomicAdd(o + 1, v.y);
    unsafeAtomicAdd(o + 2, v.z);
    unsafeAtomicAdd(o + 3, v.w);
}

// ---------------------------------------------------------------------------
// Kernel 3: in-place ReLU on the aggregated output.
// ---------------------------------------------------------------------------
__global__ __launch_bounds__(256) void gcn_relu(float* __restrict__ out, int n)
{
    const int i = blockIdx.x * blockDim.x + threadIdx.x;
    if (i < n) out[i] = fmaxf(out[i], 0.0f);
}

// ---------------------------------------------------------------------------
extern "C" void kernel_launch(void* const* d_in, const int* in_sizes, int n_in,
                              void* d_out, int out_size, void* d_ws, size_t ws_size,
                              hipStream_t stream)
{
    const float* X   = (const float*)d_in[0];   // feature [N,128]
    const float* W   = (const float*)d_in[1];   // weight  [128,64]
    const float* deg = (const float*)d_in[2];   // degree  [N]
    const int*   src = (const int*)d_in[3];     // src_idx [E]
    const int*   dst = (const int*)d_in[4];     // dst_idx [E]

    float* out = (float*)d_out;                 // [N,64]
    float* HS  = (float*)d_ws;                  // scratch: scaled h, [N,64] = 25.6 MB

    const int n_nodes = in_sizes[2];
    const int n_edges = in_sizes[3];

    // Accumulator must start at zero every call (graph replays don't re-poison).
    hipMemsetAsync(d_out, 0, (size_t)out_size * sizeof(float), stream);

    const int gemm_blocks = (n_nodes + 31) / 32;
    gcn_gemm_wmma<<<gemm_blocks, 256, 0, stream>>>(X, W, deg, HS, n_nodes);

    const long long nwork = (long long)n_edges * 16;
    const int sb = (int)((nwork + 255) / 256);
    gcn_edge_scatter<<<sb, 256, 0, stream>>>(HS, src, dst, out, n_edges);

    const int rb = (out_size + 255) / 256;
    gcn_relu<<<rb, 256, 0, stream>>>(out, out_size);
}